// GNNModel_6459630813849
// MI455X (gfx1250) — compile-verified
//
#include <hip/hip_runtime.h>

typedef __attribute__((ext_vector_type(2))) float v2f;
typedef __attribute__((ext_vector_type(8))) float v8f;

#define IN_DIM  128
#define HID_DIM 64
#define OUT_DIM 3

__device__ __forceinline__ void atomAddF(float* p, float v) {
    __hip_atomic_fetch_add(p, v, __ATOMIC_RELAXED, __HIP_MEMORY_SCOPE_AGENT);
}

// ---------------- degree / normalization ----------------
__global__ void k_deg_init(float* __restrict__ deg, int n) {
    int i = blockIdx.x * blockDim.x + threadIdx.x;
    if (i < n) deg[i] = 1.0f;                       // self-loop contribution
}

__global__ void k_deg_scatter(const int* __restrict__ dst, float* __restrict__ deg, int e) {
    int i = blockIdx.x * blockDim.x + threadIdx.x;
    if (i < e) atomAddF(&deg[dst[i]], 1.0f);
}

__global__ void k_dinv(float* __restrict__ deg, int n) {
    int i = blockIdx.x * blockDim.x + threadIdx.x;
    if (i < n) deg[i] = rsqrtf(fmaxf(deg[i], 1.0f));
}

// ---------------- layer-1 GEMM: H1 = X @ W1 (fp32 WMMA) ----------------
// One wave -> 16 rows x 64 cols. A fragment shared across the four 16-wide N tiles.
__global__ __launch_bounds__(256) void k_gemm1(const float* __restrict__ X,
                                               const float* __restrict__ W,
                                               float* __restrict__ H, int n) {
    __shared__ float sW[IN_DIM * HID_DIM];          // 32 KB of 320 KB
    for (int i = threadIdx.x; i < IN_DIM * HID_DIM; i += 256) sW[i] = W[i];
    __syncthreads();

    const int lane = threadIdx.x & 31;
    const int wave = threadIdx.x >> 5;
    const int rowBase = (blockIdx.x * 8 + wave) * 16;
    if (rowBase >= n) return;                       // wave-uniform: EXEC stays all-ones

    const int m  = lane & 15;                       // row (A) / col (B,C)
    const int kh = lane >> 4;                       // K half: 0 -> K=0,1 ; 1 -> K=2,3
    int row  = rowBase + m;
    int rowc = row < n ? row : n - 1;               // clamp tail loads
    const float* xrow = X + (size_t)rowc * IN_DIM;

    v8f a0 = {}, a1 = {}, a2 = {}, a3 = {};
#pragma unroll 4
    for (int k = 0; k < IN_DIM; k += 4) {
        v2f a = *(const v2f*)(xrow + k + 2 * kh);   // K=2kh, 2kh+1 of this row
        const float* wp = sW + (k + 2 * kh) * HID_DIM;
        v2f b0 = { wp[m +  0], wp[HID_DIM + m +  0] };
        v2f b1 = { wp[m + 16], wp[HID_DIM + m + 16] };
        v2f b2 = { wp[m + 32], wp[HID_DIM + m + 32] };
        v2f b3 = { wp[m + 48], wp[HID_DIM + m + 48] };
        a0 = __builtin_amdgcn_wmma_f32_16x16x4_f32(false, a, false, b0, (short)0, a0, false, false);
        a1 = __builtin_amdgcn_wmma_f32_16x16x4_f32(false, a, false, b1, (short)0, a1, false, false);
        a2 = __builtin_amdgcn_wmma_f32_16x16x4_f32(false, a, false, b2, (short)0, a2, false, false);
        a3 = __builtin_amdgcn_wmma_f32_16x16x4_f32(false, a, false, b3, (short)0, a3, false, false);
    }
#pragma unroll
    for (int r = 0; r < 8; ++r) {                   // C/D: VGPR r -> M = r + 8*kh, N = m
        int orow = rowBase + r + 8 * kh;
        if (orow < n) {
            float* hp = H + (size_t)orow * HID_DIM + m;
            hp[0]  = a0[r];
            hp[16] = a1[r];
            hp[32] = a2[r];
            hp[48] = a3[r];
        }
    }
}

// ---------------- layer-1 self-loop + bias init: A1 = dinv^2 * H1 + b1 ----------------
__global__ void k_self1(const float* __restrict__ H1, const float* __restrict__ dinv,
                        const float* __restrict__ b1, float* __restrict__ A1, int n) {
    int idx = blockIdx.x * blockDim.x + threadIdx.x;
    if (idx < n * HID_DIM) {
        int i = idx / HID_DIM, c = idx % HID_DIM;
        float w = dinv[i];
        A1[idx] = H1[idx] * w * w + b1[c];
    }
}

// ---------------- layer-1 edge aggregation (32 lanes per edge, 2 ch/lane) ----------------
__global__ void k_agg1(const int* __restrict__ src, const int* __restrict__ dst,
                       const float* __restrict__ dinv, const float* __restrict__ H1,
                       float* __restrict__ A1, int e) {
    long long idx = (long long)blockIdx.x * blockDim.x + threadIdx.x;
    if (idx >= (long long)e * 32) return;
    int ed = (int)(idx >> 5);
    int l  = (int)(idx & 31);
    int s = src[ed], d = dst[ed];
    float w = dinv[s] * dinv[d];
    v2f h = *(const v2f*)(H1 + (size_t)s * HID_DIM + l * 2);
    float* op = A1 + (size_t)d * HID_DIM + l * 2;
    atomAddF(op + 0, h.x * w);
    atomAddF(op + 1, h.y * w);
}

// ---------------- layer-2 GEMM: H2 = relu(A1) @ W2 (fp32 WMMA, N padded to 16) ----------------
__global__ __launch_bounds__(256) void k_gemm2(const float* __restrict__ A1,
                                               const float* __restrict__ W2,
                                               float* __restrict__ H2, int n) {
    const int lane = threadIdx.x & 31;
    const int wave = threadIdx.x >> 5;
    const int rowBase = (blockIdx.x * 8 + wave) * 16;
    if (rowBase >= n) return;

    const int m  = lane & 15;
    const int kh = lane >> 4;
    int row  = rowBase + m;
    int rowc = row < n ? row : n - 1;
    const float* arow = A1 + (size_t)rowc * HID_DIM;

    v8f acc = {};
#pragma unroll 4
    for (int k = 0; k < HID_DIM; k += 4) {
        v2f a = *(const v2f*)(arow + k + 2 * kh);
        a.x = fmaxf(a.x, 0.0f);                     // ReLU folded into A load
        a.y = fmaxf(a.y, 0.0f);
        v2f b = { 0.0f, 0.0f };
        if (m < OUT_DIM) {
            b.x = W2[(k + 2 * kh) * OUT_DIM + m];
            b.y = W2[(k + 2 * kh + 1) * OUT_DIM + m];
        }
        acc = __builtin_amdgcn_wmma_f32_16x16x4_f32(false, a, false, b, (short)0, acc, false, false);
    }
#pragma unroll
    for (int r = 0; r < 8; ++r) {
        int orow = rowBase + r + 8 * kh;
        if (orow < n && m < OUT_DIM) H2[(size_t)orow * OUT_DIM + m] = acc[r];
    }
}

// ---------------- layer-2 self-loop + bias init: out = dinv^2 * H2 + b2 ----------------
__global__ void k_self2(const float* __restrict__ H2, const float* __restrict__ dinv,
                        const float* __restrict__ b2, float* __restrict__ out, int n) {
    int idx = blockIdx.x * blockDim.x + threadIdx.x;
    if (idx < n * OUT_DIM) {
        int i = idx / OUT_DIM, c = idx % OUT_DIM;
        float w = dinv[i];
        out[idx] = H2[idx] * w * w + b2[c];
    }
}

// ---------------- layer-2 edge aggregation (1 thread per edge, 3 channels) ----------------
__global__ void k_agg2(const int* __restrict__ src, const int* __restrict__ dst,
                       const float* __restrict__ dinv, const float* __restrict__ H2,
                       float* __restrict__ out, int e) {
    int i = blockIdx.x * blockDim.x + threadIdx.x;
    if (i < e) {
        int s = src[i], d = dst[i];
        float w = dinv[s] * dinv[d];
        const float* hp = H2 + (size_t)s * OUT_DIM;
        float* op = out + (size_t)d * OUT_DIM;
        atomAddF(op + 0, hp[0] * w);
        atomAddF(op + 1, hp[1] * w);
        atomAddF(op + 2, hp[2] * w);
    }
}

extern "C" void kernel_launch(void* const* d_in, const int* in_sizes, int n_in,
                              void* d_out, int out_size, void* d_ws, size_t ws_size,
                              hipStream_t stream) {
    (void)n_in; (void)out_size; (void)ws_size;
    const float* x  = (const float*)d_in[0];
    const int*  edge = (const int*)d_in[1];
    const float* W1 = (const float*)d_in[2];
    const float* b1 = (const float*)d_in[3];
    const float* W2 = (const float*)d_in[4];
    const float* b2 = (const float*)d_in[5];
    float* out = (float*)d_out;

    const int n = in_sizes[0] / IN_DIM;     // 100000
    const int e = in_sizes[1] / 2;          // 3200000
    const int* src = edge;
    const int* dst = edge + e;

    // workspace layout (floats): dinv[n] | H1[n*64] | A1[n*64] | H2[n*3]  (~53 MB)
    float* ws   = (float*)d_ws;
    float* dinv = ws;
    float* H1   = ws + n;
    float* A1   = H1 + (size_t)n * HID_DIM;
    float* H2   = A1 + (size_t)n * HID_DIM;

    dim3 blk(256);
    k_deg_init   <<<(n + 255) / 256, blk, 0, stream>>>(dinv, n);
    k_deg_scatter<<<(e + 255) / 256, blk, 0, stream>>>(dst, dinv, e);
    k_dinv       <<<(n + 255) / 256, blk, 0, stream>>>(dinv, n);

    k_gemm1<<<(n + 127) / 128, blk, 0, stream>>>(x, W1, H1, n);
    k_self1<<<(n * HID_DIM + 255) / 256, blk, 0, stream>>>(H1, dinv, b1, A1, n);
    long long work1 = (long long)e * 32;
    k_agg1<<<(unsigned)((work1 + 255) / 256), blk, 0, stream>>>(src, dst, dinv, H1, A1, e);

    k_gemm2<<<(n + 127) / 128, blk, 0, stream>>>(A1, W2, H2, n);
    k_self2<<<(n * OUT_DIM + 255) / 256, blk, 0, stream>>>(H2, dinv, b2, out, n);
    k_agg2<<<(e + 255) / 256, blk, 0, stream>>>(src, dst, dinv, H2, out, e);
}